// PerturbationEvolver_71305047048605
// MI455X (gfx1250) — compile-verified
//
#include <hip/hip_runtime.h>
#include <math.h>

// ---------------- problem constants (match reference) ----------------
#define NKM      2048
#define DST      12
#define NSTEPS   1500
#define NOUT     500

__constant__ const float kLNA0 = -14.0f;
__constant__ const float kLNA1 = 0.0f;

constexpr float H0f       = 2.25e-4f;
constexpr float OM_Mf     = 0.31f;
constexpr float OM_Bf     = 0.049f;
constexpr float OM_Rf     = 8.4e-5f;
constexpr float OM_Cf     = (float)(0.31 - 0.049);
constexpr float OM_Lf     = (float)(1.0 - 0.31 - 8.4e-5);
constexpr float RNUf      = 0.405f;
constexpr float KAPPA0f   = 1e-4f;
constexpr float KAPPA_CAPf= 80.0f;
constexpr float KH_CAPf   = 150.0f;
constexpr float CS2f      = 1e-5f;

// ---------------- background coefficients (lna-only, lane-uniform) ----------------
struct Coef {
  float aHv, inv_aH2, Rc, Rb, Rg, kappa, coupb;
};

__device__ __forceinline__ Coef make_coef(float lna) {
  Coef c;
  float a     = expf(lna);
  float inva  = 1.0f / a;
  float inva2 = inva * inva;
  float E2    = OM_Rf * inva2 * inva2 + OM_Mf * inva2 * inva + OM_Lf;
  c.aHv     = H0f * a * sqrtf(E2);
  c.inv_aH2 = 1.0f / (c.aHv * c.aHv);
  const float f = 1.5f * H0f * H0f;
  c.Rc    = f * OM_Cf * inva;
  c.Rb    = f * OM_Bf * inva;
  c.Rg    = f * OM_Rf * inva2;
  c.kappa = fminf(KAPPA0f * inva2, KAPPA_CAPf);
  c.coupb = fminf(c.kappa * (4.0f * c.Rg) / (3.0f * c.Rb), KAPPA_CAPf);
  return c;
}

// ---------------- per-mode derivative (synchronous-gauge structure) ----------------
__device__ __forceinline__ void deriv(const Coef& c, float k,
                                      const float* __restrict__ y,
                                      float* __restrict__ dy) {
  float kh     = fminf(k / c.aHv, KH_CAPf);
  float inv_aH = kh / k;
  float eta = y[1];
  float d_c = y[2], d_b = y[3], th_b = y[4];
  float d_g = y[5], th_g = y[6], sig  = y[7];
  float F3  = y[8], G0  = y[9], G1   = y[10], G2 = y[11];

  float Sd  = c.Rc * d_c + c.Rb * d_b + c.Rg * d_g;
  float St  = c.Rb * th_b + (4.0f / 3.0f) * c.Rg * th_g;
  float hp  = 2.0f * (kh * kh * eta + Sd * c.inv_aH2);
  float etap = St * inv_aH / (k * k);
  float Pol  = sig + G0 + G2;

  dy[0] = hp;
  dy[1] = etap;
  dy[2] = -0.5f * hp;
  dy[3] = -th_b * inv_aH - 0.5f * hp;
  dy[4] = -th_b + CS2f * k * kh * d_b + c.coupb * (th_g - th_b);
  dy[5] = -(4.0f / 3.0f) * th_g * inv_aH - (2.0f / 3.0f) * hp;
  dy[6] = k * kh * (0.25f * d_g - sig) + c.kappa * (th_b - th_g);
  dy[7] = (4.0f / 15.0f) * th_g * inv_aH + (2.0f / 15.0f) * hp
        + (4.0f / 5.0f) * etap - (3.0f / 10.0f) * kh * F3
        - c.kappa * (sig - 0.1f * Pol);
  dy[8]  = (6.0f / 7.0f) * kh * sig - c.kappa * F3;
  dy[9]  = -kh * G1 - c.kappa * (G0 - 0.5f * Pol);
  dy[10] = (kh / 3.0f) * (G0 - 2.0f * G2) - c.kappa * G1;
  dy[11] = (2.0f / 5.0f) * kh * G1 - c.kappa * (G2 - 0.1f * Pol);
}

// ---------------- RK4 integrator: one thread per mode, state in VGPRs ----------------
// traj != nullptr : store full trajectory [1501, 12, NKM] into workspace (L2-resident)
// outp != nullptr : fused scalar interpolation straight into d_out (fallback path)
__global__ void __launch_bounds__(32)
evolve_kernel(const float* __restrict__ k_log, const float* __restrict__ amp_in,
              float* __restrict__ traj, float* __restrict__ outp) {
  int m = blockIdx.x * blockDim.x + threadIdx.x;
  if (m >= NKM) return;

  // k = 1e-4 * 4000^k_log
  float k  = 1e-4f * expf(k_log[m] * logf(4000.0f));
  float am = amp_in[m];

  // adiabatic ICs at lna = LNA0
  Coef ci = make_coef(kLNA0);
  float tau_ini = 1.0f / ci.aHv;
  float kt  = k * tau_ini;
  float kt2 = kt * kt;
  float h0   = 0.5f * kt2 * am;
  float eta0 = am * (1.0f - (5.0f / (12.0f * (15.0f + 4.0f * RNUf))) * kt2);
  float d_g0 = -(2.0f / 3.0f) * h0;
  float d_m0 = 0.75f * d_g0;
  float th0  = -(k / 18.0f) * kt2 * kt * am;

  float y[DST] = {h0, eta0, d_m0, d_m0, th0, d_g0, th0,
                  0.0f, 0.0f, 0.0f, 0.0f, 0.0f};
  float yprev[DST];
#pragma unroll
  for (int t = 0; t < DST; ++t) yprev[t] = y[t];

  if (traj) {
#pragma unroll
    for (int t = 0; t < DST; ++t) traj[t * NKM + m] = y[t];
  }

  const float dl = (kLNA1 - kLNA0) / (float)NSTEPS;
  int jOut = 0;

  for (int i = 0; i < NSTEPS; ++i) {
    float lna = kLNA0 + dl * (float)i;
    Coef c0 = make_coef(lna);
    Coef cm = make_coef(lna + 0.5f * dl);
    Coef c1 = make_coef(lna + dl);

    float k1v[DST], k2v[DST], k3v[DST], k4v[DST], yt[DST];
    deriv(c0, k, y, k1v);
#pragma unroll
    for (int t = 0; t < DST; ++t) yt[t] = y[t] + 0.5f * dl * k1v[t];
    deriv(cm, k, yt, k2v);
#pragma unroll
    for (int t = 0; t < DST; ++t) yt[t] = y[t] + 0.5f * dl * k2v[t];
    deriv(cm, k, yt, k3v);
#pragma unroll
    for (int t = 0; t < DST; ++t) yt[t] = y[t] + dl * k3v[t];
    deriv(c1, k, yt, k4v);
#pragma unroll
    for (int t = 0; t < DST; ++t)
      y[t] = y[t] + (dl / 6.0f) * (k1v[t] + 2.0f * k2v[t] + 2.0f * k3v[t] + k4v[t]);

    if (traj) {
#pragma unroll
      for (int t = 0; t < DST; ++t)
        traj[((i + 1) * DST + t) * NKM + m] = y[t];
    }

    if (outp) {
      // emit all output samples whose lower knot is step i (uniform across wave)
      while (jOut < NOUT) {
        float p  = (float)(jOut * NSTEPS) / (float)(NOUT - 1);
        int   ib = (int)p;
        if (ib > NSTEPS - 1) ib = NSTEPS - 1;
        if (ib != i) break;
        float w = p - (float)ib;
#pragma unroll
        for (int t = 0; t < DST; ++t)
          outp[(t * NOUT + jOut) * NKM + m] = yprev[t] + w * (y[t] - yprev[t]);
        ++jOut;
      }
    }
#pragma unroll
    for (int t = 0; t < DST; ++t) yprev[t] = y[t];
  }
}

// ---------------- WMMA resampling: OUT_d[500x2048] = W[500x1501] * TRAJ_d[1501x2048] ----
// Banded GEMM via V_WMMA_F32_16X16X4_F32; 13 chained K=4 WMMAs cover the <=48-row band
// of any 16-row output tile. TRAJ (147.5 MB) is L2-resident on MI455X (192 MB L2).
typedef float v2f __attribute__((ext_vector_type(2)));
typedef float v8f __attribute__((ext_vector_type(8)));

__global__ void __launch_bounds__(32)
interp_wmma_kernel(const float* __restrict__ traj, float* __restrict__ out) {
  const int mt   = blockIdx.x;          // mode tile  (128)
  const int jt   = blockIdx.y;          // out-row tile (32)
  const int d    = blockIdx.z;          // field (12)
  const int lane = threadIdx.x;         // 0..31, one wave per block
  const int half = lane >> 4;           // K-half select
  const int lidx = lane & 15;           // M row (A) / N col (B)

  const int j0 = jt * 16;
  const int m0 = mt * 16;

  // uniform lowest knot touched by this row tile
  float p0 = (float)(j0 * NSTEPS) / (float)(NOUT - 1);
  int iMin = (int)p0; if (iMin > NSTEPS - 1) iMin = NSTEPS - 1;

  // this lane's A row = output sample j0+lidx -> (knot ib, weight w)
  const int jrow = j0 + lidx;
  float pj = (float)(jrow * NSTEPS) / (float)(NOUT - 1);
  int   ib = (int)pj; if (ib > NSTEPS - 1) ib = NSTEPS - 1;
  float w  = pj - (float)ib;
  const bool jvalid = (jrow < NOUT);

  v8f acc = {0.0f, 0.0f, 0.0f, 0.0f, 0.0f, 0.0f, 0.0f, 0.0f};

  const int NCHUNK = 13;                 // 13*4 = 52 >= band width 48
  for (int c = 0; c < NCHUNK; ++c) {
    int kb = iMin + c * 4 + half * 2;    // K indices {kb, kb+1} for this lane

    // A fragment: interpolation weights W[jrow, kb], W[jrow, kb+1]
    float wx = (jvalid && kb     == ib) ? (1.0f - w)
             : (jvalid && kb     == ib + 1) ? w : 0.0f;
    float wy = (jvalid && kb + 1 == ib) ? (1.0f - w)
             : (jvalid && kb + 1 == ib + 1) ? w : 0.0f;
    v2f a; a.x = wx; a.y = wy;

    // B fragment: TRAJ rows kb, kb+1, column m0+lidx (clamped loads x 0 weight)
    int r0 = kb;     if (r0 > NSTEPS) r0 = NSTEPS;
    int r1 = kb + 1; if (r1 > NSTEPS) r1 = NSTEPS;
    v2f b;
    b.x = traj[(r0 * DST + d) * NKM + (m0 + lidx)];
    b.y = traj[(r1 * DST + d) * NKM + (m0 + lidx)];

    acc = __builtin_amdgcn_wmma_f32_16x16x4_f32(
        /*neg_a=*/false, a, /*neg_b=*/false, b,
        /*c_mod=*/(short)0, acc, /*reuse_a=*/false, /*reuse_b=*/false);
  }

  // D layout: VGPR g -> row g + 8*half, col = lidx
#pragma unroll
  for (int g = 0; g < 8; ++g) {
    int j = j0 + g + 8 * half;
    if (j < NOUT)
      out[(d * NOUT + j) * NKM + (m0 + lidx)] = acc[g];
  }
}

// ---------------- host side ----------------
extern "C" void kernel_launch(void* const* d_in, const int* in_sizes, int n_in,
                              void* d_out, int out_size, void* d_ws, size_t ws_size,
                              hipStream_t stream) {
  (void)in_sizes; (void)n_in; (void)out_size;
  const float* k_log = (const float*)d_in[0];
  const float* amp   = (const float*)d_in[1];
  float*       out   = (float*)d_out;

  const size_t trajBytes = (size_t)(NSTEPS + 1) * DST * NKM * sizeof(float); // 147.5 MB

  if (ws_size >= trajBytes) {
    float* traj = (float*)d_ws;
    // integrate, stream trajectory into workspace (L2-resident)
    evolve_kernel<<<dim3(NKM / 32), dim3(32), 0, stream>>>(k_log, amp, traj, nullptr);
    // banded-GEMM resampling on the matrix pipes
    dim3 grid(NKM / 16, (NOUT + 15) / 16, DST);
    interp_wmma_kernel<<<grid, dim3(32), 0, stream>>>(traj, out);
  } else {
    // fallback: fused scalar interpolation, no scratch required
    evolve_kernel<<<dim3(NKM / 32), dim3(32), 0, stream>>>(k_log, amp, nullptr, out);
  }
}